// RoiPoolingConvSingle_24773371363472
// MI455X (gfx1250) — compile-verified
//
#include <hip/hip_runtime.h>
#include <math.h>

// ---------------------------------------------------------------------------
// RoIPool (single ROI, 7x7 bins, max-pool, clamp-to-0) for a 200x200x1024 f32
// feature map (NHWC). Pure bandwidth-bound streaming max-reduce:
//   ~76 MB read once @ 23.3 TB/s  ->  ~3.3 us floor on MI455X.
// wave32 mapping: 32 lanes x float4 = 512B coalesced global_load_b128 per wave,
// non-temporal (read-once stream, keep the 192MB L2 clean), global_prefetch_b8
// for the next row-group, LDS reduction across the 8 row-group waves, and the
// output written via the CDNA5 async path (global_store_async_from_lds_b128,
// ASYNCcnt) since the reduced tile already lives in LDS.
// ---------------------------------------------------------------------------

#define POOL_H 7
#define POOL_W 7
#define H_FEAT 200
#define W_FEAT 200
#define C_FEAT 1024
#define ROWGRP 8   // row-groups per bin (threadIdx.y)

typedef float v4f __attribute__((ext_vector_type(4)));
typedef int   v4i __attribute__((ext_vector_type(4)));

#if __has_builtin(__builtin_amdgcn_global_store_async_from_lds_b128)
#define USE_ASYNC_OUT 1
#else
#define USE_ASYNC_OUT 0
#endif

__device__ __forceinline__ v4f vmax4(v4f a, v4f b) {
    v4f r;
    r.x = fmaxf(a.x, b.x);
    r.y = fmaxf(a.y, b.y);
    r.z = fmaxf(a.z, b.z);
    r.w = fmaxf(a.w, b.w);
    return r;
}

__global__ __launch_bounds__(256)
void roipool_max_kernel(const float* __restrict__ img,   // [1,200,200,1024]
                        const float* __restrict__ roi,   // [1,5]
                        float* __restrict__ out)         // [1,7,7,1024]
{
    const int tx    = threadIdx.x;   // 0..31 : channel quad lane
    const int ty    = threadIdx.y;   // 0..7  : row group
    const int chunk = blockIdx.x;    // 0..7  : 128-channel chunk
    const int pw    = blockIdx.y;    // 0..6
    const int ph    = blockIdx.z;    // 0..6

    // ---- decode ROI exactly like the reference (float32 ops, rint = jnp.round)
    const float r1 = roi[1], r2 = roi[2], r3 = roi[3], r4 = roi[4];
    const int roi_start_w = (int)(rintf(r1) * 1.0f);
    const int roi_start_h = (int)(rintf(r2) * 1.0f);
    const int roi_end_w   = (int)(rintf(r3) * 1.0f);
    const int roi_end_h   = (int)(rintf(r4) * 1.0f);

    const int roi_h = max(roi_end_h - roi_start_h + 1, 1);
    const int roi_w = max(roi_end_w - roi_start_w + 1, 1);
    const float bin_h = (float)roi_h / (float)POOL_H;
    const float bin_w = (float)roi_w / (float)POOL_W;

    int hstart = (int)floorf((float)ph * bin_h)          + roi_start_h;
    int hend   = (int)ceilf (((float)ph + 1.0f) * bin_h) + roi_start_h;
    int wstart = (int)floorf((float)pw * bin_w)          + roi_start_w;
    int wend   = (int)ceilf (((float)pw + 1.0f) * bin_w) + roi_start_w;
    hstart = min(max(hstart, 0), H_FEAT);
    hend   = min(max(hend,   0), H_FEAT);
    wstart = min(max(wstart, 0), W_FEAT);
    wend   = min(max(wend,   0), W_FEAT);

    const int c4 = (chunk * 32 + tx) * 4;    // first of this lane's 4 channels

    v4f m;
    m.x = -INFINITY; m.y = -INFINITY; m.z = -INFINITY; m.w = -INFINITY;

    const int wstride4 = C_FEAT / 4;         // v4f elements between adjacent w

    for (int h = hstart + ty; h < hend; h += ROWGRP) {
        const v4f* p = (const v4f*)(img + (size_t)(h * W_FEAT + wstart) * C_FEAT + c4);

        // CDNA5: global_prefetch_b8 for next row-group of this lane
        if (h + ROWGRP < hend) {
            __builtin_prefetch(
                (const void*)(img + (size_t)((h + ROWGRP) * W_FEAT + wstart) * C_FEAT + c4),
                0, 0);
        }

        int w = wstart;
        for (; w + 4 <= wend; w += 4) {      // 4 outstanding b128 NT loads
            v4f a = __builtin_nontemporal_load(p + 0 * wstride4);
            v4f b = __builtin_nontemporal_load(p + 1 * wstride4);
            v4f c = __builtin_nontemporal_load(p + 2 * wstride4);
            v4f d = __builtin_nontemporal_load(p + 3 * wstride4);
            m = vmax4(m, vmax4(vmax4(a, b), vmax4(c, d)));
            p += 4 * wstride4;
        }
        for (; w < wend; ++w) {
            m = vmax4(m, __builtin_nontemporal_load(p));
            p += wstride4;
        }
    }

    // ---- cross-row-group reduction through LDS (ds_store/ds_load)
    __shared__ v4f red[ROWGRP][32];
    red[ty][tx] = m;
    __syncthreads();

    if (ty == 0) {
        v4f v = red[0][tx];
#pragma unroll
        for (int i = 1; i < ROWGRP; ++i) v = vmax4(v, red[i][tx]);
        // top_data starts at 0 in the reference -> clamp (also covers empty bins)
        v.x = fmaxf(v.x, 0.0f);
        v.y = fmaxf(v.y, 0.0f);
        v.z = fmaxf(v.z, 0.0f);
        v.w = fmaxf(v.w, 0.0f);

        float* op = out + (size_t)((ph * POOL_W + pw) * C_FEAT) + c4;
#if USE_ASYNC_OUT
        // Final tile already in LDS: ship it out on the CDNA5 async path.
        red[0][tx] = v;                                   // ds_store_b128
        asm volatile("s_wait_dscnt 0x0" ::: "memory");    // LDS write visible to async engine
        __builtin_amdgcn_global_store_async_from_lds_b128(
            (v4i*)op, (v4i*)&red[0][tx], 0, 0);
        asm volatile("s_wait_asynccnt 0x0" ::: "memory"); // drain before wave end
#else
        *(v4f*)op = v;
#endif
    }
}

extern "C" void kernel_launch(void* const* d_in, const int* in_sizes, int n_in,
                              void* d_out, int out_size, void* d_ws, size_t ws_size,
                              hipStream_t stream) {
    (void)in_sizes; (void)n_in; (void)out_size; (void)d_ws; (void)ws_size;
    const float* img = (const float*)d_in[0];   // [1,200,200,1024] f32
    const float* roi = (const float*)d_in[1];   // [1,5] f32
    float* out = (float*)d_out;                 // [1,7,7,1024] f32

    dim3 grid(C_FEAT / 128, POOL_W, POOL_H);    // (8, 7, 7) = 392 blocks
    dim3 block(32, ROWGRP, 1);                  // 256 threads = 8 wave32
    roipool_max_kernel<<<grid, block, 0, stream>>>(img, roi, out);
}